// GAT_85658827751998
// MI455X (gfx1250) — compile-verified
//
#include <hip/hip_runtime.h>
#include <hip/hip_bf16.h>
#include <math.h>

// ---------------------------------------------------------------------------
// GAT forward for MI455X (gfx1250).
//   * fp32 WMMA (v_wmma_f32_16x16x4_f32) for all dense GEMMs (exact fp32).
//   * Edge phase: streaming passes with L2-resident float atomics; the whole
//     node feature working set (~77 MB) fits in the 192 MB L2.
//   * Aggregation: one wave32 per edge, lane l owns channels [4l,4l+4) ->
//     every gather is a coalesced global_load_b128, scatter is 4x atomic_add.
// ---------------------------------------------------------------------------

typedef float v2f __attribute__((ext_vector_type(2)));
typedef float v8f __attribute__((ext_vector_type(8)));

#define F_IN    256
#define HID     32
#define HEADS   4
#define HC      128   // HID*HEADS
#define CLASSES 40
#define LAYERS  2
#define SLOPE   0.1f

// ---------------------------------------------------------------------------
// C[M x 128] = A[M x K] * B[K x 128], fp32 WMMA.
// grid.x = M/16 (M must be a multiple of 16; N=50000 = 3125*16).
// block = 256 threads = 8 waves; wave w computes cols [16w, 16w+16).
// ---------------------------------------------------------------------------
__global__ __launch_bounds__(256)
void gemm_f32_wmma(const float* __restrict__ A, const float* __restrict__ B,
                   float* __restrict__ C, int M, int K) {
  const int lane = threadIdx.x & 31;
  const int wave = threadIdx.x >> 5;
  const int row0 = blockIdx.x * 16;
  const int col0 = wave * 16;
  const int half = lane >> 4;   // 0: lanes 0-15, 1: lanes 16-31
  const int l16  = lane & 15;

  const float* __restrict__ Arow = A + (size_t)(row0 + l16) * K;

  v8f acc = {};
  for (int k0 = 0; k0 < K; k0 += 4) {
    // A 16x4 frag: VGPR0 = K(k0 | k0+2), VGPR1 = K(k0+1 | k0+3)
    v2f a, b;
    a.x = Arow[k0 + 2 * half + 0];
    a.y = Arow[k0 + 2 * half + 1];
    // B 4x16 frag (row striped across lanes, halves carry K pairs)
    b.x = B[(size_t)(k0 + 2 * half + 0) * HC + col0 + l16];
    b.y = B[(size_t)(k0 + 2 * half + 1) * HC + col0 + l16];
    acc = __builtin_amdgcn_wmma_f32_16x16x4_f32(
        /*neg_a=*/false, a, /*neg_b=*/false, b,
        /*c_mod=*/(short)0, acc, /*reuse_a=*/false, /*reuse_b=*/false);
  }
  // C/D layout: VGPR v -> row (v + 8*half), col l16
#pragma unroll
  for (int v = 0; v < 8; ++v)
    C[(size_t)(row0 + v + 8 * half) * HC + col0 + l16] = acc[v];
}

// ---------------------------------------------------------------------------
__global__ void fill_f32(float* __restrict__ p, float v, int n) {
  int t = blockIdx.x * blockDim.x + threadIdx.x;
  if (t < n) p[t] = v;
}

// alpha_src/alpha_dst per (node, head): dot(hp[n,h,:], att[h,:HID] / att[h,HID:])
__global__ void node_alpha(const float* __restrict__ hp,
                           const float* __restrict__ att,   // [HEADS][2*HID]
                           float* __restrict__ asrc, float* __restrict__ adst,
                           int N) {
  int t = blockIdx.x * blockDim.x + threadIdx.x;
  if (t >= N * HEADS) return;
  int n = t / HEADS, hd = t % HEADS;
  const float* v  = hp + (size_t)n * HC + hd * HID;
  const float* as = att + hd * 2 * HID;
  const float* ad = as + HID;
  float s0 = 0.f, s1 = 0.f;
#pragma unroll 8
  for (int c = 0; c < HID; ++c) { s0 += v[c] * as[c]; s1 += v[c] * ad[c]; }
  asrc[t] = s0;
  adst[t] = s1;
}

// float atomic max via monotonic bit-pattern trick (init with -inf)
__device__ __forceinline__ void atomicMaxF(float* addr, float val) {
  if (val >= 0.0f)
    atomicMax((int*)addr, __float_as_int(val));
  else
    atomicMin((unsigned int*)addr, __float_as_uint(val));
}

// Pass 1: logits = leaky_relu(asrc[src]+adst[dst]); store + segment max.
__global__ void edge_logits_max(const int* __restrict__ ei, int E, int Etot,
                                const float* __restrict__ asrc,
                                const float* __restrict__ adst,
                                float* __restrict__ elog, float* __restrict__ m) {
  int e = blockIdx.x * blockDim.x + threadIdx.x;
  if (e >= Etot) return;
  int s, d;
  if (e < E) { s = ei[e]; d = ei[E + e]; } else { s = d = e - E; }
  const float4 a = *(const float4*)(asrc + (size_t)s * HEADS);
  const float4 b = *(const float4*)(adst + (size_t)d * HEADS);
  float lg[HEADS] = {a.x + b.x, a.y + b.y, a.z + b.z, a.w + b.w};
  float* mrow = m + (size_t)d * HEADS;
#pragma unroll
  for (int hd = 0; hd < HEADS; ++hd) {
    float v = lg[hd] > 0.f ? lg[hd] : SLOPE * lg[hd];
    elog[(size_t)e * HEADS + hd] = v;
    atomicMaxF(mrow + hd, v);
  }
}

// Pass 2: e = exp(logit - m[dst]); store; segment sum into z.
__global__ void edge_exp_sum(const int* __restrict__ ei, int E, int Etot,
                             const float* __restrict__ m, float* __restrict__ elog,
                             float* __restrict__ z, const float* __restrict__ hp) {
  int e = blockIdx.x * blockDim.x + threadIdx.x;
  if (e >= Etot) return;
  int s, d;
  if (e < E) { s = ei[e]; d = ei[E + e]; } else { s = d = e - E; }
  // warm L2/L0 for the aggregation pass gather of hp[src]
  __builtin_prefetch(hp + (size_t)s * HC, 0, 0);
  const float4 mm = *(const float4*)(m + (size_t)d * HEADS);
  float4 lg = *(const float4*)(elog + (size_t)e * HEADS);
  float4 ev = {__expf(lg.x - mm.x), __expf(lg.y - mm.y),
               __expf(lg.z - mm.z), __expf(lg.w - mm.w)};
  *(float4*)(elog + (size_t)e * HEADS) = ev;
  float* zr = z + (size_t)d * HEADS;
  atomicAdd(zr + 0, ev.x);
  atomicAdd(zr + 1, ev.y);
  atomicAdd(zr + 2, ev.z);
  atomicAdd(zr + 3, ev.w);
}

// Pass 3: one wave per edge; lane l owns channels [4l,4l+4) (single head each).
// agg[dst] += hp[src] * alpha
__global__ __launch_bounds__(256)
void edge_aggregate(const int* __restrict__ ei, int E, int Etot,
                    const float* __restrict__ hp, const float* __restrict__ elog,
                    const float* __restrict__ z, float* __restrict__ agg) {
  long long gid = (long long)blockIdx.x * blockDim.x + threadIdx.x;
  int e    = (int)(gid >> 5);
  int lane = (int)(gid & 31);
  if (e >= Etot) return;
  int s, d;
  if (e < E) { s = ei[e]; d = ei[E + e]; } else { s = d = e - E; }
  int hd = lane >> 3;   // 4 channels per lane all belong to one head
  float alpha = elog[(size_t)e * HEADS + hd] / z[(size_t)d * HEADS + hd];
  const float4 hv = *(const float4*)(hp + (size_t)s * HC + lane * 4);
  float* dp = agg + (size_t)d * HC + lane * 4;
  atomicAdd(dp + 0, hv.x * alpha);
  atomicAdd(dp + 1, hv.y * alpha);
  atomicAdd(dp + 2, hv.z * alpha);
  atomicAdd(dp + 3, hv.w * alpha);
}

// h = elu(agg + bias)
__global__ void node_bias_elu(const float* __restrict__ agg,
                              const float* __restrict__ bias,
                              float* __restrict__ h, int n) {
  int t = blockIdx.x * blockDim.x + threadIdx.x;
  if (t >= n) return;
  float v = agg[t] + bias[t & (HC - 1)];
  h[t] = v > 0.f ? v : __expf(v) - 1.f;
}

// out[n,c] = dot(h[n,:], Wout[:,c])  (128x40, tiny -> VALU)
__global__ void out_gemm(const float* __restrict__ h, const float* __restrict__ Wout,
                         float* __restrict__ out, int N) {
  int t = blockIdx.x * blockDim.x + threadIdx.x;
  if (t >= N * CLASSES) return;
  int n = t / CLASSES, c = t % CLASSES;
  const float* hr = h + (size_t)n * HC;
  float s = 0.f;
#pragma unroll 8
  for (int k = 0; k < HC; ++k) s += hr[k] * Wout[(size_t)k * CLASSES + c];
  out[t] = s;
}

__global__ void log_softmax_rows(float* __restrict__ out, int N) {
  int n = blockIdx.x * blockDim.x + threadIdx.x;
  if (n >= N) return;
  float* row = out + (size_t)n * CLASSES;
  float mx = row[0];
#pragma unroll
  for (int c = 1; c < CLASSES; ++c) mx = fmaxf(mx, row[c]);
  float sum = 0.f;
#pragma unroll
  for (int c = 0; c < CLASSES; ++c) sum += __expf(row[c] - mx);
  float lse = mx + __logf(sum);
#pragma unroll
  for (int c = 0; c < CLASSES; ++c) row[c] -= lse;
}

// ---------------------------------------------------------------------------
extern "C" void kernel_launch(void* const* d_in, const int* in_sizes, int n_in,
                              void* d_out, int out_size, void* d_ws, size_t ws_size,
                              hipStream_t stream) {
  const float* x     = (const float*)d_in[0];
  const int*   eidx  = (const int*)d_in[1];     // [2][E] (JAX default int32)
  const float* Win   = (const float*)d_in[2];   // [F_IN][HC]
  const float* Ws    = (const float*)d_in[3];   // [LAYERS][HC][HC]
  const float* atts  = (const float*)d_in[4];   // [LAYERS][HEADS][2*HID]
  const float* bias  = (const float*)d_in[5];   // [LAYERS][HC]
  const float* Wout  = (const float*)d_in[6];   // [HC][CLASSES]
  float*       out   = (float*)d_out;

  const int N    = in_sizes[0] / F_IN;          // 50000 (multiple of 16)
  const int E    = in_sizes[1] / 2;             // 800000
  const int Etot = E + N;                       // edges + self loops

  // workspace carve-up (~94 MB of f32)
  float* ws   = (float*)d_ws;
  float* h    = ws; ws += (size_t)N * HC;
  float* hp   = ws; ws += (size_t)N * HC;
  float* agg  = ws; ws += (size_t)N * HC;
  float* asrc = ws; ws += (size_t)N * HEADS;
  float* adst = ws; ws += (size_t)N * HEADS;
  float* mbuf = ws; ws += (size_t)N * HEADS;
  float* zbuf = ws; ws += (size_t)N * HEADS;
  float* elog = ws; ws += (size_t)Etot * HEADS;

  const int TB = 256;
  const int gNH  = (N * HEADS + TB - 1) / TB;
  const int gNC  = (N * HC + TB - 1) / TB;
  const int gE   = (Etot + TB - 1) / TB;
  const int gE32 = (int)(((long long)Etot * 32 + TB - 1) / TB);

  // h = x @ Win   (K = F_IN)
  gemm_f32_wmma<<<N / 16, TB, 0, stream>>>(x, Win, h, N, F_IN);

  for (int l = 0; l < LAYERS; ++l) {
    const float* Wl  = Ws + (size_t)l * HC * HC;
    const float* al  = atts + (size_t)l * HEADS * 2 * HID;
    const float* bl  = bias + (size_t)l * HC;

    // hp = h @ Wl  (K = HC)
    gemm_f32_wmma<<<N / 16, TB, 0, stream>>>(h, Wl, hp, N, HC);

    node_alpha<<<gNH, TB, 0, stream>>>(hp, al, asrc, adst, N);

    fill_f32<<<gNH, TB, 0, stream>>>(mbuf, -INFINITY, N * HEADS);
    fill_f32<<<gNH, TB, 0, stream>>>(zbuf, 0.0f, N * HEADS);
    fill_f32<<<gNC, TB, 0, stream>>>(agg, 0.0f, N * HC);

    edge_logits_max<<<gE, TB, 0, stream>>>(eidx, E, Etot, asrc, adst, elog, mbuf);
    edge_exp_sum<<<gE, TB, 0, stream>>>(eidx, E, Etot, mbuf, elog, zbuf, hp);
    edge_aggregate<<<gE32, TB, 0, stream>>>(eidx, E, Etot, hp, elog, zbuf, agg);

    node_bias_elu<<<gNC, TB, 0, stream>>>(agg, bl, h, N * HC);
  }

  out_gemm<<<(N * CLASSES + TB - 1) / TB, TB, 0, stream>>>(h, Wout, out, N);
  log_softmax_rows<<<(N + TB - 1) / TB, TB, 0, stream>>>(out, N);
}